// SparseCodingEmbedding_42863773614863
// MI455X (gfx1250) — compile-verified
//
#include <hip/hip_runtime.h>
#include <stdint.h>

// Problem constants (from the reference)
#define ROWS_   65536
#define DIM_    64
#define NCH_    8
#define BS_     131072
#define VOCAB_  1000000

#define THREADS 256
#define WPB     8      // waves per block (wave32)
#define ITERS_  16     // batch elements per wave (== WMMA M dimension)
#define ROWF    80     // padded LDS row stride in floats (bank-conflict-free)
#define BUF_F   (NCH_ * ROWF)   // 640 floats per staging buffer

typedef __attribute__((ext_vector_type(2))) float v2f;
typedef __attribute__((ext_vector_type(8))) float v8f;

// ---------- monotonic f32 <-> uint encoding for atomic min/max ----------
__device__ __forceinline__ unsigned f2ord(float f) {
  unsigned u = __float_as_uint(f);
  return (u & 0x80000000u) ? ~u : (u | 0x80000000u);
}
__device__ __forceinline__ float ord2f(unsigned u) {
  return __uint_as_float((u & 0x80000000u) ? (u ^ 0x80000000u) : ~u);
}

// deterministic per-element uniform(0,1) for stochastic rounding
__device__ __forceinline__ float hash01(unsigned v) {
  v ^= v >> 16; v *= 0x7feb352du;
  v ^= v >> 15; v *= 0x846ca68bu;
  v ^= v >> 16;
  return (float)(v >> 8) * (1.0f / 16777216.0f);
}

__global__ void init_ws(unsigned* ws) { ws[0] = 0xFFFFFFFFu; ws[1] = 0u; }

// ---------- global min/max over weights (8M f32, ~1.4us at 23.3 TB/s) ----------
__global__ void __launch_bounds__(THREADS)
minmax_kernel(const float4* __restrict__ w4, int n4, unsigned* __restrict__ ws) {
  float mn = 3.4e38f, mx = -3.4e38f;
  for (int i = blockIdx.x * blockDim.x + threadIdx.x; i < n4;
       i += gridDim.x * blockDim.x) {
    float4 v = w4[i];
    mn = fminf(mn, fminf(fminf(v.x, v.y), fminf(v.z, v.w)));
    mx = fmaxf(mx, fmaxf(fmaxf(v.x, v.y), fmaxf(v.z, v.w)));
  }
#pragma unroll
  for (int m = 16; m >= 1; m >>= 1) {          // wave32 reduction
    mn = fminf(mn, __shfl_xor(mn, m, 32));
    mx = fmaxf(mx, __shfl_xor(mx, m, 32));
  }
  __shared__ float smn[WPB], smx[WPB];
  int lane = threadIdx.x & 31, wv = threadIdx.x >> 5;
  if (lane == 0) { smn[wv] = mn; smx[wv] = mx; }
  __syncthreads();
  if (threadIdx.x == 0) {
#pragma unroll
    for (int i = 1; i < WPB; ++i) { mn = fminf(mn, smn[i]); mx = fmaxf(mx, smx[i]); }
    atomicMin(&ws[0], f2ord(mn));
    atomicMax(&ws[1], f2ord(mx));
  }
}

// ---------- main gather + quantize + WMMA chunk-sum ----------
__global__ void __launch_bounds__(THREADS)
gather_kernel(const float* __restrict__ table, const float* __restrict__ weights,
              const int* __restrict__ h, const int* __restrict__ x,
              const unsigned* __restrict__ ws, float* __restrict__ out) {
  // double-buffered staging: 8 waves * 2 bufs * (8 rows * 80 f32) = 40 KB
  __shared__ __align__(16) float lbuf[WPB * 2 * BUF_F];

  const int lane   = threadIdx.x & 31;
  const int wv     = threadIdx.x >> 5;
  const int waveId = blockIdx.x * WPB + wv;
  const int b0     = waveId * ITERS_;

  // quantization parameters (uniform scalar loads; one div at entry only)
  float mn = ord2f(ws[0]);
  float mx = ord2f(ws[1]);
  float scale = (mx - mn) * (1.0f / 255.0f);
  if (!(scale > 0.0f)) scale = 1.0f;
  const float rscale = 1.0f / scale;          // hoisted reciprocal
  const float zp = -128.0f - mn * rscale;

  // lanes 0..15 hold x[b0..b0+15]
  int xall = (lane < ITERS_) ? x[b0 + lane] : 0;

  // prefetch the dependent h / weights rows for all 16 batch elements
#pragma unroll
  for (int i = 0; i < ITERS_; ++i) {
    int xv = __shfl(xall, i, 32);
    if (lane == 0) {
      __builtin_prefetch(h + (size_t)xv * NCH_);
      __builtin_prefetch(weights + (size_t)xv * NCH_);
    }
  }

  const int c = lane & 7;   // chunk owned by this lane (replicated x4 across wave)

  auto load_meta = [&](int xv, int& hc, float& wq) {
    size_t ofs = (size_t)xv * NCH_ + c;
    hc = h[ofs];
    float w  = weights[ofs];
    float q  = fmaf(w, rscale, zp);          // single FMA on the hot path
    float fq = floorf(q);
    float pr = q - fq;                       // P(round up)
    float r  = hash01((unsigned)ofs * 2654435761u + 0x9e3779b9u);
    float qi = fq + ((r < pr) ? 1.0f : 0.0f);
    qi = fminf(fmaxf(qi, -128.0f), 127.0f);  // int8 clamp
    wq = scale * (qi - zp);                  // dequantize
  };

  const unsigned ldsByteBase =
      (unsigned)(uintptr_t)lbuf + (unsigned)wv * (2u * BUF_F * 4u);
  const int half = lane >> 4;   // which row of a pair this lane fetches / K offset
  const int l16  = lane & 15;
  const int kof  = half * 2;    // per-lane K offset for both A and B operands

  // gather the 8 table rows for one batch element into LDS, asynchronously:
  // 4x global_load_async_to_lds_b128, each instruction moves 2 full 256B rows.
  auto issue = [&](int hc, int buf) {
    unsigned base = ldsByteBase + (unsigned)buf * (BUF_F * 4u);
#pragma unroll
    for (int j = 0; j < 4; ++j) {
      int row = 2 * j + half;
      int idx = __shfl(hc, row, 32);
      unsigned long long gp =
          (unsigned long long)(uintptr_t)(table + (size_t)idx * DIM_ + l16 * 4);
      unsigned loff = base + (unsigned)row * (ROWF * 4u) + (unsigned)l16 * 16u;
      asm volatile("global_load_async_to_lds_b128 %0, %1, off"
                   :: "v"(loff), "v"(gp) : "memory");
    }
  };

  // 16x64 output block accumulated in 4 WMMA tiles (exact f32 math)
  v8f zero8 = {0.f, 0.f, 0.f, 0.f, 0.f, 0.f, 0.f, 0.f};
  v8f acc[4];
#pragma unroll
  for (int t = 0; t < 4; ++t) acc[t] = zero8;

  int   hc, hcn;
  float wq, wqn;
  load_meta(__shfl(xall, 0, 32), hc, wq);
  issue(hc, 0);

  int cur = 0;
  for (int i = 0; i < ITERS_; ++i) {
    if (i + 1 < ITERS_) {
      // software pipeline: start next element's gather before consuming current
      load_meta(__shfl(xall, i + 1, 32), hcn, wqn);
      issue(hcn, cur ^ 1);
      asm volatile("s_wait_asynccnt 4" ::: "memory");  // current buffer done
    } else {
      asm volatile("s_wait_asynccnt 0" ::: "memory");
    }

    const float* lb = lbuf + (wv * 2 + cur) * BUF_F;
    const bool sel = (l16 == i);  // A nonzero only in M-row == batch slot i

#pragma unroll
    for (int g = 0; g < 2; ++g) {         // two K-groups cover the 8 chunks
      const int k0 = 4 * g + kof;
      const float w0 = __shfl(wq, k0, 32);
      const float w1 = __shfl(wq, k0 + 1, 32);
      v2f a;                               // A 16x4: row i = 4 chunk weights
      a.x = sel ? w0 : 0.0f;
      a.y = sel ? w1 : 0.0f;
      const float* r0 = lb + k0 * ROWF + l16;  // B 4x16: gathered table rows
      const float* r1 = r0 + ROWF;
#pragma unroll
      for (int t = 0; t < 4; ++t) {        // 4 dim tiles of 16
        v2f bb;
        bb.x = r0[16 * t];
        bb.y = r1[16 * t];
        acc[t] = __builtin_amdgcn_wmma_f32_16x16x4_f32(
            false, a, false, bb, (short)0, acc[t], false, false);
      }
    }

    hc = hcn; wq = wqn; cur ^= 1;
  }

  // store the 16x64 block: VGPR v of D holds rows v (lanes 0-15) / v+8 (16-31)
  float* op = out + (size_t)b0 * DIM_;
#pragma unroll
  for (int v = 0; v < 8; ++v) {
    float* row = op + (size_t)(v + 8 * half) * DIM_ + l16;
#pragma unroll
    for (int t = 0; t < 4; ++t) row[16 * t] = acc[t][v];
  }
}

extern "C" void kernel_launch(void* const* d_in, const int* in_sizes, int n_in,
                              void* d_out, int out_size, void* d_ws, size_t ws_size,
                              hipStream_t stream) {
  (void)in_sizes; (void)n_in; (void)out_size; (void)ws_size;
  const float* table   = (const float*)d_in[0];
  const float* weights = (const float*)d_in[1];
  const int*   h       = (const int*)d_in[2];
  const int*   x       = (const int*)d_in[3];
  unsigned*    ws      = (unsigned*)d_ws;
  float*       out     = (float*)d_out;

  init_ws<<<1, 1, 0, stream>>>(ws);

  int n4 = (VOCAB_ * NCH_) / 4;  // 2M float4
  minmax_kernel<<<4096, THREADS, 0, stream>>>((const float4*)weights, n4, ws);

  gather_kernel<<<BS_ / (WPB * ITERS_), THREADS, 0, stream>>>(
      table, weights, h, x, ws, out);
}